// ChamferVAELoss_47493748359776
// MI455X (gfx1250) — compile-verified
//
#include <hip/hip_runtime.h>
#include <stdint.h>

// Problem constants (from reference): B=16, N=1024, D=8, output scalar.
#define BB     16
#define NN     1024
#define DDIM   8
#define SPLIT  16             // pred-row splits per batch
#define TSPLIT 4              // target-col splits per batch
#define TPB    64             // threads per block (2 wave32s); NN/SPLIT == TPB
#define JTILE  (NN / TSPLIT)  // 256 targets per block
#define INF_BITS 0x7F800000   // +inf as int; non-negative floats order like ints

#if defined(__gfx1250__) && \
    __has_builtin(__builtin_amdgcn_global_load_async_to_lds_b128) && \
    __has_builtin(__builtin_amdgcn_s_wait_asynccnt)
#define USE_ASYNC_LDS 1
#endif

#if defined(USE_ASYNC_LDS)
typedef int v4i __attribute__((ext_vector_type(4)));

__device__ __forceinline__ void async_copy_b128(const void* g, void* l) {
  // Params are generic v4i*; codegen inserts the AS1/AS3 casts.
  __builtin_amdgcn_global_load_async_to_lds_b128(
      (v4i*)g, (v4i*)l, /*offset=*/0, /*cpol=*/0);
}
#endif

__device__ __forceinline__ float smooth_l1(float d) {
  float ad = fabsf(d);
  return (ad < 1.0f) ? (0.5f * d * d) : (ad - 0.5f);
}

// Workspace layout: minArr[0 .. B*N)        = per-target mins (int bits)
//                   minArr[B*N .. 2*B*N)    = per-pred  mins (int bits)

// ---------------------------------------------------------------- init
__global__ void chamfer_init(int* __restrict__ minArr) {
  int i = blockIdx.x * blockDim.x + threadIdx.x;
  if (i < 2 * BB * NN) minArr[i] = INF_BITS;
}

// ---------------------------------------------------------------- main
__global__ __launch_bounds__(TPB) void chamfer_main(
    const float* __restrict__ pred, const float* __restrict__ realp,
    int* __restrict__ minArr) {
  __shared__ float realS[JTILE * DDIM];  // 8 KB: real[b, j0:j0+256, :]
  __shared__ int   minTgtLoc[JTILE];     // 1 KB

  const int t   = threadIdx.x;
  const int b   = blockIdx.x / (SPLIT * TSPLIT);
  const int rem = blockIdx.x % (SPLIT * TSPLIT);
  const int s   = rem / TSPLIT;
  const int ts  = rem % TSPLIT;
  const int r   = s * TPB + t;       // this thread's pred row
  const int j0  = ts * JTILE;        // this block's target range start

  for (int j = t; j < JTILE; j += TPB) minTgtLoc[j] = INF_BITS;

  // Stage real[b, j0:j0+JTILE, :] (512 x float4) into LDS.
  const float4* src = (const float4*)(realp + ((size_t)b * NN + j0) * DDIM);
  float4*       dst = (float4*)realS;
#if defined(USE_ASYNC_LDS)
  #pragma unroll
  for (int k = 0; k < (JTILE * DDIM / 4) / TPB; ++k) {
    int idx = t + k * TPB;
    async_copy_b128(src + idx, dst + idx);
  }
  __builtin_amdgcn_s_wait_asynccnt(0);
#else
  for (int k = 0; k < (JTILE * DDIM / 4) / TPB; ++k) {
    int idx = t + k * TPB;
    dst[idx] = src[idx];
  }
#endif
  __syncthreads();

  // Own pred row in registers.
  const float4* op = (const float4*)(pred + ((size_t)b * NN + r) * DDIM);
  float4 x0 = op[0], x1 = op[1];
  float xr[8] = {x0.x, x0.y, x0.z, x0.w, x1.x, x1.y, x1.z, x1.w};

  const int lane = t & 31;
  float minPred = __int_as_float(INF_BITS);

  #pragma unroll 2
  for (int j = 0; j < JTILE; ++j) {
    // Uniform j across the wave -> LDS broadcast reads, conflict-free.
    float4 r0 = ((const float4*)realS)[j * 2 + 0];
    float4 r1 = ((const float4*)realS)[j * 2 + 1];
    float rr[8] = {r0.x, r0.y, r0.z, r0.w, r1.x, r1.y, r1.z, r1.w};

    float acc = 0.0f;
    #pragma unroll
    for (int d = 0; d < 8; ++d) acc += smooth_l1(xr[d] - rr[d]);
    float se = 0.125f * acc;             // mean over D

    minPred = fminf(minPred, se);        // min over targets (axis=2 term)

    // Wave32 min across the 32 pred rows of this wave -> one LDS atomic per j.
    float w = se;
    #pragma unroll
    for (int off = 16; off > 0; off >>= 1)
      w = fminf(w, __shfl_xor(w, off, 32));
    if (lane == 0) atomicMin(&minTgtLoc[j], __float_as_int(w));
  }

  // Per-pred min: combine across the TSPLIT target tiles.
  atomicMin(&minArr[BB * NN + b * NN + r], __float_as_int(minPred));

  __syncthreads();
  // Per-target mins: combine across the SPLIT pred tiles.
  for (int j = t; j < JTILE; j += TPB)
    atomicMin(&minArr[b * NN + j0 + j], minTgtLoc[j]);
}

// ---------------------------------------------------------------- finish
__global__ __launch_bounds__(256) void chamfer_finish(
    const int* __restrict__ minArr, float* __restrict__ out) {
  __shared__ float red[256];
  const int t = threadIdx.x;
  float s = 0.0f;
  for (int i = t; i < 2 * BB * NN; i += 256) s += __int_as_float(minArr[i]);
  red[t] = s;
  __syncthreads();
  for (int k = 128; k > 0; k >>= 1) {
    if (t < k) red[t] += red[t + k];
    __syncthreads();
  }
  // 0.0 * mean(dkl) contributes exactly 0 for finite inputs.
  if (t == 0) out[0] = red[0] * (1.0f / (float)(BB * NN));
}

extern "C" void kernel_launch(void* const* d_in, const int* in_sizes, int n_in,
                              void* d_out, int out_size, void* d_ws, size_t ws_size,
                              hipStream_t stream) {
  const float* pred  = (const float*)d_in[0];  // "output" [B,N,D]
  // d_in[1]=mu, d_in[2]=log_var: multiplied by 0.0 in the reference -> unused.
  const float* realp = (const float*)d_in[3];  // "real"   [B,N,D]
  float* outd = (float*)d_out;

  int* minArr = (int*)d_ws;  // 2 * B * N ints = 128 KB

  chamfer_init<<<(2 * BB * NN + 255) / 256, 256, 0, stream>>>(minArr);
  chamfer_main<<<BB * SPLIT * TSPLIT, TPB, 0, stream>>>(pred, realp, minArr);
  chamfer_finish<<<1, 256, 0, stream>>>(minArr, outd);
}